// NodeBlock_74234214744860
// MI455X (gfx1250) — compile-verified
//
#include <hip/hip_runtime.h>
#include <hip/hip_bf16.h>

#define N_NODES 50000
#define N_EDGES 1600000
#define D_EDGE  64
#define D_NODE  64
#define D_HID   256
#define D_OUT   64
#define N_TILES (N_NODES / 16)   // 3125, exact

typedef __attribute__((ext_vector_type(2))) float v2f;
typedef __attribute__((ext_vector_type(4))) float v4f;
typedef __attribute__((ext_vector_type(8))) float v8f;

// ---------------------------------------------------------------- zero ws
__global__ void zero_kernel(float4* __restrict__ p, int n4) {
    int i = blockIdx.x * blockDim.x + threadIdx.x;
    if (i < n4) p[i] = make_float4(0.f, 0.f, 0.f, 0.f);
}

// ------------------------------------------------- segment-sum via atomics
// one thread per (edge, 4-column group); 16 groups of float4 per edge.
// edge_feats is streamed once -> non-temporal loads keep L2 free for the
// hot 12.8MB atomic destination region.
__global__ __launch_bounds__(256)
void scatter_kernel(const float* __restrict__ ef, const int* __restrict__ recv,
                    float* __restrict__ agg, float* __restrict__ cnt) {
    unsigned gid = blockIdx.x * blockDim.x + threadIdx.x;
    if (gid >= (unsigned)N_EDGES * 16u) return;
    int e  = gid >> 4;
    int c4 = gid & 15;
    int r  = recv[e];
    v4f v = __builtin_nontemporal_load((const v4f*)ef + gid);  // ef[e*64+c4*4..]
    float* dst = agg + (size_t)r * D_EDGE + c4 * 4;
    unsafeAtomicAdd(dst + 0, v.x);
    unsafeAtomicAdd(dst + 1, v.y);
    unsafeAtomicAdd(dst + 2, v.z);
    unsafeAtomicAdd(dst + 3, v.w);
    if (c4 == 0) unsafeAtomicAdd(cnt + r, 1.0f);
}

// ---------------------------------------------------------------- MLP
__device__ __forceinline__ v8f wmma4(v2f a, v2f b, v8f c) {
    return __builtin_amdgcn_wmma_f32_16x16x4_f32(
        /*neg_a=*/false, a, /*neg_b=*/false, b,
        /*c_mod=*/(short)0, c, /*reuse_a=*/false, /*reuse_b=*/false);
}

#define MLP_WAVES 4
// dynamic LDS layout (floats): W1[32768] | W2[16384] | X[4][2048] | H[4][4096]
#define SM_W2 32768
#define SM_X  (SM_W2 + 16384)
#define SM_H  (SM_X + MLP_WAVES * 2048)
#define SMEM_FLOATS (SM_H + MLP_WAVES * 4096)       // 73728 floats = 288 KB
#define SMEM_BYTES  (SMEM_FLOATS * 4)

__global__ __launch_bounds__(MLP_WAVES * 32)
void mlp_kernel(const float* __restrict__ nf, const float* __restrict__ agg,
                const float* __restrict__ cnt,
                const float* __restrict__ W1, const float* __restrict__ B1,
                const float* __restrict__ W2, const float* __restrict__ B2,
                float* __restrict__ out) {
    extern __shared__ float smem[];
    float* sW1 = smem;
    float* sW2 = smem + SM_W2;
    const int tid  = threadIdx.x;
    const int wave = tid >> 5;
    const int lane = tid & 31;
    float* sX = smem + SM_X + wave * 2048;          // 16 x 128 per wave
    float* sH = smem + SM_H + wave * 4096;          // 16 x 256 per wave

    // stage weights once per block (float4 vectorized)
    for (int i = tid; i < (128 * 256) / 4; i += blockDim.x)
        ((float4*)sW1)[i] = ((const float4*)W1)[i];
    for (int i = tid; i < (256 * 64) / 4; i += blockDim.x)
        ((float4*)sW2)[i] = ((const float4*)W2)[i];
    __syncthreads();

    const int row_in = lane & 15;      // M (or N) index for frags
    const int hi     = lane >> 4;      // half-wave select
    const int kh     = hi * 2;         // K sub-offset within a K-block of 4

    for (int t = blockIdx.x * MLP_WAVES + wave; t < N_TILES;
         t += gridDim.x * MLP_WAVES) {
        const int r0 = t * 16;

        // ---- build x tile in LDS: cols 0-63 = agg/cnt mean, 64-127 = node
        for (int j = 0; j < 16; ++j) {
            const int rg = r0 + j;
            float4 v;
            if (lane < 16) {
                v = ((const float4*)(agg + (size_t)rg * D_EDGE))[lane];
                float s = 1.0f / fmaxf(cnt[rg], 1.0f);
                v.x *= s; v.y *= s; v.z *= s; v.w *= s;
            } else {
                v = ((const float4*)(nf + (size_t)rg * D_NODE))[lane - 16];
            }
            ((float4*)(sX + j * 128))[lane] = v;
        }
        asm volatile("s_wait_dscnt 0" ::: "memory");   // wave-local LDS RAW

        // ---- GEMM1: h[16x256] = relu(x @ W1 + b1), 4 n-tiles per chunk
        for (int ntc = 0; ntc < 4; ++ntc) {
            v8f a0 = {}, a1 = {}, a2 = {}, a3 = {};
            const int n0 = ntc * 64 + row_in;
            for (int kb = 0; kb < 32; ++kb) {
                const int k0 = kb * 4 + kh;
                // contiguous even K-pair -> single ds_load_b64
                v2f a = *(const v2f*)(sX + row_in * 128 + k0);
                const float* w = sW1 + k0 * 256 + n0;
                v2f b0; b0.x = w[0];  b0.y = w[256];
                v2f b1; b1.x = w[16]; b1.y = w[16 + 256];
                v2f b2; b2.x = w[32]; b2.y = w[32 + 256];
                v2f b3; b3.x = w[48]; b3.y = w[48 + 256];
                a0 = wmma4(a, b0, a0);
                a1 = wmma4(a, b1, a1);
                a2 = wmma4(a, b2, a2);
                a3 = wmma4(a, b3, a3);
            }
            #pragma unroll
            for (int u = 0; u < 4; ++u) {
                const int col = n0 + u * 16;
                const float bias = B1[col];
                v8f acc = (u == 0) ? a0 : (u == 1) ? a1 : (u == 2) ? a2 : a3;
                #pragma unroll
                for (int j = 0; j < 8; ++j)
                    sH[(j + 8 * hi) * 256 + col] = fmaxf(acc[j] + bias, 0.0f);
            }
        }
        asm volatile("s_wait_dscnt 0" ::: "memory");   // h visible to wave

        // ---- GEMM2: out[16x64] = h @ W2 + b2 (all 4 n-tiles at once)
        {
            v8f a0 = {}, a1 = {}, a2 = {}, a3 = {};
            for (int kb = 0; kb < 64; ++kb) {
                const int k0 = kb * 4 + kh;
                v2f a = *(const v2f*)(sH + row_in * 256 + k0);
                const float* w = sW2 + k0 * 64 + row_in;
                v2f b0; b0.x = w[0];  b0.y = w[64];
                v2f b1; b1.x = w[16]; b1.y = w[16 + 64];
                v2f b2; b2.x = w[32]; b2.y = w[32 + 64];
                v2f b3; b3.x = w[48]; b3.y = w[48 + 64];
                a0 = wmma4(a, b0, a0);
                a1 = wmma4(a, b1, a1);
                a2 = wmma4(a, b2, a2);
                a3 = wmma4(a, b3, a3);
            }
            #pragma unroll
            for (int u = 0; u < 4; ++u) {
                const int col = u * 16 + row_in;
                const float bias = B2[col];
                v8f acc = (u == 0) ? a0 : (u == 1) ? a1 : (u == 2) ? a2 : a3;
                #pragma unroll
                for (int j = 0; j < 8; ++j)
                    __builtin_nontemporal_store(
                        acc[j] + bias,
                        &out[(size_t)(r0 + j + 8 * hi) * D_OUT + col]);
            }
        }
    }
}

// ---------------------------------------------------------------- host
extern "C" void kernel_launch(void* const* d_in, const int* in_sizes, int n_in,
                              void* d_out, int out_size, void* d_ws, size_t ws_size,
                              hipStream_t stream) {
    const float* node_feats = (const float*)d_in[0];
    const float* edge_feats = (const float*)d_in[1];
    const int*   receivers  = (const int*)d_in[2];
    const float* W1 = (const float*)d_in[3];
    const float* b1 = (const float*)d_in[4];
    const float* W2 = (const float*)d_in[5];
    const float* b2 = (const float*)d_in[6];
    float* out = (float*)d_out;

    float* agg = (float*)d_ws;                        // 50000*64 f32
    float* cnt = agg + (size_t)N_NODES * D_EDGE;      // 50000   f32

    // 1) zero accumulators
    const int n4 = (N_NODES * D_EDGE + N_NODES) / 4;  // 812500 float4
    zero_kernel<<<(n4 + 255) / 256, 256, 0, stream>>>((float4*)d_ws, n4);

    // 2) segment-sum scatter (bandwidth-critical: ~410 MB streamed)
    const unsigned sc_threads = (unsigned)N_EDGES * 16u;
    scatter_kernel<<<(sc_threads + 255) / 256, 256, 0, stream>>>(
        edge_feats, receivers, agg, cnt);

    // 3) mean + concat + MLP (WMMA f32)
    (void)hipFuncSetAttribute((const void*)mlp_kernel,
                              hipFuncAttributeMaxDynamicSharedMemorySize,
                              SMEM_BYTES);
    mlp_kernel<<<256, MLP_WAVES * 32, SMEM_BYTES, stream>>>(
        node_feats, agg, cnt, W1, b1, W2, b2, out);
}